// CircuitModule_18236431139024
// MI455X (gfx1250) — compile-verified
//
#include <hip/hip_runtime.h>

typedef int   v4i __attribute__((ext_vector_type(4)));
typedef float v2f __attribute__((ext_vector_type(2)));

// ---------------------------------------------------------------------------
// Leaf decode, fused into layer 0 (never materialize the [0,1,p,1-p,...] vec):
//   x[0]=0, x[1]=1, x[2+2k]=pos[k], x[3+2k]=1-pos[k]
// Branchless: unconditional (clamped) gather + selects, so no EXEC divergence.
// ---------------------------------------------------------------------------
__device__ __forceinline__ float decode_leaf(const float* __restrict__ xp, int idx) {
    int   j = idx - 2;
    int   k = (j < 0 ? 0 : j) >> 1;     // clamp so the gather is always in-bounds
    float p = xp[k];                    // L2-resident gather (x_pos = 8 MB << 192 MB L2)
    float v = (j & 1) ? (1.0f - p) : p;
    if (j < 0) v = (float)idx;          // idx 0 -> 0.0, idx 1 -> 1.0
    return v;
}

// ---------------------------------------------------------------------------
// Layer 0: binary product over leaf decodes. 2 products / thread so the index
// stream is one 128-bit NT load per thread (16 B, perfectly coalesced).
// ---------------------------------------------------------------------------
__global__ void prod_leaf_kernel(const float* __restrict__ xp,
                                 const int*   __restrict__ ix,   // 2*m indices
                                 float*       __restrict__ out,  // m products
                                 int m) {
    int t = blockIdx.x * blockDim.x + threadIdx.x;
    int i = t * 2;
    if (i + 1 < m) {
        v4i q = __builtin_nontemporal_load((const v4i*)(ix + 2 * i)); // global_load_b128 ... th:NT
        float a = decode_leaf(xp, q.x) * decode_leaf(xp, q.y);
        float b = decode_leaf(xp, q.z) * decode_leaf(xp, q.w);
        v2f r; r.x = a; r.y = b;
        *(v2f*)(out + i) = r;           // l0 stays temporal: re-read by layer-1 gathers via L2
    } else if (i < m) {
        int a0 = ix[2 * i], a1 = ix[2 * i + 1];
        out[i] = decode_leaf(xp, a0) * decode_leaf(xp, a1);
    }
}

// ---------------------------------------------------------------------------
// Layer 2: binary product gathering from the previous sum layer.
// ---------------------------------------------------------------------------
__global__ void prod_kernel(const float* __restrict__ in,
                            const int*   __restrict__ ix,   // 2*m indices
                            float*       __restrict__ out,  // m products
                            int m) {
    int t = blockIdx.x * blockDim.x + threadIdx.x;
    int i = t * 2;
    if (i + 1 < m) {
        v4i q = __builtin_nontemporal_load((const v4i*)(ix + 2 * i));
        float a = in[q.x] * in[q.y];    // gathers hit L2 (l1 = 4 MB)
        float b = in[q.z] * in[q.w];
        v2f r; r.x = a; r.y = b;
        *(v2f*)(out + i) = r;
    } else if (i < m) {
        out[i] = in[ix[2 * i]] * in[ix[2 * i + 1]];
    }
}

// ---------------------------------------------------------------------------
// Segment sum with SORTED segment ids. Each thread owns 8 contiguous edges
// (avg run length is exactly 8 for both sum layers), accumulates runs in
// registers, and emits a non-returning global_atomic_add_f32 only at run
// boundaries: ~8x fewer L2 atomics than naive scatter-add.
// Index streams: 4x 128-bit NT loads per thread.
// ---------------------------------------------------------------------------
__device__ __forceinline__ void seg_flush(float* __restrict__ out, int seg, float acc) {
    (void)__hip_atomic_fetch_add(out + seg, acc, __ATOMIC_RELAXED,
                                 __HIP_MEMORY_SCOPE_AGENT);  // global_atomic_add_f32 (no return)
}

__global__ void segsum8_kernel(const float* __restrict__ src,
                               const int*   __restrict__ ix_in,
                               const int*   __restrict__ ix_out,
                               float*       __restrict__ out,
                               int n_edges) {
    int t    = blockIdx.x * blockDim.x + threadIdx.x;
    int base = t * 8;
    if (base >= n_edges) return;

    if (base + 8 <= n_edges) {
        v4i si0 = __builtin_nontemporal_load((const v4i*)(ix_in  + base));
        v4i si1 = __builtin_nontemporal_load((const v4i*)(ix_in  + base + 4));
        v4i so0 = __builtin_nontemporal_load((const v4i*)(ix_out + base));
        v4i so1 = __builtin_nontemporal_load((const v4i*)(ix_out + base + 4));
        int srcs[8] = { si0.x, si0.y, si0.z, si0.w, si1.x, si1.y, si1.z, si1.w };
        int segs[8] = { so0.x, so0.y, so0.z, so0.w, so1.x, so1.y, so1.z, so1.w };

        int   cur = segs[0];
        float acc = 0.0f;
#pragma unroll
        for (int k = 0; k < 8; ++k) {
            float v = src[srcs[k]];     // L2-resident gather
            if (segs[k] != cur) { seg_flush(out, cur, acc); cur = segs[k]; acc = 0.0f; }
            acc += v;
        }
        seg_flush(out, cur, acc);
    } else {
        int   cur = ix_out[base];
        float acc = 0.0f;
        for (int k = base; k < n_edges; ++k) {
            int   seg = ix_out[k];
            float v   = src[ix_in[k]];
            if (seg != cur) { seg_flush(out, cur, acc); cur = seg; acc = 0.0f; }
            acc += v;
        }
        seg_flush(out, cur, acc);
    }
}

// ---------------------------------------------------------------------------
// Zero-init for atomic accumulation targets (re-run every call: the harness
// does not re-poison between graph replays, and atomics need clean zeros).
// ---------------------------------------------------------------------------
__global__ void zero_kernel(float* __restrict__ p, int n) {
    int i = blockIdx.x * blockDim.x + threadIdx.x;
    if (i < n) p[i] = 0.0f;
}

static inline unsigned int cdiv(long a, long b) { return (unsigned int)((a + b - 1) / b); }

extern "C" void kernel_launch(void* const* d_in, const int* in_sizes, int n_in,
                              void* d_out, int out_size, void* d_ws, size_t ws_size,
                              hipStream_t stream) {
    const float* x_pos   = (const float*)d_in[0];
    const int*   ix_in0  = (const int*)  d_in[1];
    const int*   ix_in1  = (const int*)  d_in[2];
    const int*   ix_out1 = (const int*)  d_in[3];
    const int*   ix_in2  = (const int*)  d_in[4];
    const int*   ix_in3  = (const int*)  d_in[5];
    const int*   ix_out3 = (const int*)  d_in[6];
    float*       out     = (float*)d_out;

    const int M0 = in_sizes[1] / 2;   // 4,000,000 products in layer 0
    const int E1 = in_sizes[2];       // 8,000,000 edges into sum layer 1
    const int M2 = in_sizes[4] / 2;   //   500,000 products in layer 2
    const int E3 = in_sizes[5];       // 1,000,000 edges into sum layer 3
    const int M1 = 1000000;           // reference num_segments for layer 1 (fixed constant)
    const int M3 = out_size;          //   125,000 outputs

    // Workspace layout (all 256 B aligned): l0 | l1 | l2  (~22 MB, fits in L2)
    char*  ws = (char*)d_ws;
    size_t o0 = 0;
    size_t o1 = o0 + (((size_t)M0 * 4 + 255) & ~(size_t)255);
    size_t o2 = o1 + (((size_t)M1 * 4 + 255) & ~(size_t)255);
    float* l0 = (float*)(ws + o0);
    float* l1 = (float*)(ws + o1);
    float* l2 = (float*)(ws + o2);
    (void)ws_size; (void)n_in;

    const int B = 256;  // 8 wave32 waves per block

    // Layer 0: fused leaf-encode + binary product
    prod_leaf_kernel<<<cdiv((M0 + 1) / 2, B), B, 0, stream>>>(x_pos, ix_in0, l0, M0);

    // Layer 1: sorted segment-sum with run-coalesced float atomics
    zero_kernel<<<cdiv(M1, B), B, 0, stream>>>(l1, M1);
    segsum8_kernel<<<cdiv(cdiv(E1, 8), B), B, 0, stream>>>(l0, ix_in1, ix_out1, l1, E1);

    // Layer 2: binary product
    prod_kernel<<<cdiv((M2 + 1) / 2, B), B, 0, stream>>>(l1, ix_in2, l2, M2);

    // Layer 3: sorted segment-sum into d_out
    zero_kernel<<<cdiv(M3, B), B, 0, stream>>>(out, M3);
    segsum8_kernel<<<cdiv(cdiv(E3, 8), B), B, 0, stream>>>(l2, ix_in3, ix_out3, out, E3);
}